// AdvancedQuanvolutionFilter_65481071408818
// MI455X (gfx1250) — compile-verified
//
#include <hip/hip_runtime.h>

typedef __attribute__((ext_vector_type(2))) float v2f;
typedef __attribute__((ext_vector_type(8))) float v8f;

// ---------------- precompute: U(theta) 16x16 complex, Wt = (conv_w * Zsign)^T, bias --------
struct C2x2 { float r00,i00,r01,i01,r10,i10,r11,i11; };

__device__ inline void apply_pair(float* sr, float* si, const C2x2& U, int i0, int i1) {
  float ar = sr[i0], ai = si[i0], br = sr[i1], bi = si[i1];
  sr[i0] = U.r00*ar - U.i00*ai + U.r01*br - U.i01*bi;
  si[i0] = U.r00*ai + U.i00*ar + U.r01*bi + U.i01*br;
  sr[i1] = U.r10*ar - U.i10*ai + U.r11*br - U.i11*bi;
  si[i1] = U.r10*ai + U.i10*ar + U.r11*bi + U.i11*br;
}
__device__ inline void apply1q(float* sr, float* si, const C2x2& U, int w) {
  int tbit = 1 << (3 - w);                     // wire w -> bit (3-w) of flat index
  for (int i = 0; i < 16; ++i)
    if (!(i & tbit)) apply_pair(sr, si, U, i, i | tbit);
}
__device__ inline void applyCtrl(float* sr, float* si, const C2x2& U, int cq, int tq) {
  int cbit = 1 << (3 - cq), tbit = 1 << (3 - tq);
  for (int i = 0; i < 16; ++i)
    if ((i & cbit) && !(i & tbit)) apply_pair(sr, si, U, i, i | tbit);
}
__device__ inline C2x2 gRX(float t){ float c=cosf(0.5f*t), s=sinf(0.5f*t); return {c,0.f, 0.f,-s, 0.f,-s, c,0.f}; }
__device__ inline C2x2 gRY(float t){ float c=cosf(0.5f*t), s=sinf(0.5f*t); return {c,0.f, -s,0.f, s,0.f, c,0.f}; }
__device__ inline C2x2 gRZ(float t){ float c=cosf(0.5f*t), s=sinf(0.5f*t); return {c,-s, 0.f,0.f, 0.f,0.f, c,s}; }

// ws layout (floats): [0,256) Ur row-major U[n][k]; [256,512) Ui; [512,576) Wt[16][4]; [576,580) bias
__global__ void quanv_precompute(const float* __restrict__ theta,
                                 const float* __restrict__ conv_w,
                                 const float* __restrict__ conv_b,
                                 float* __restrict__ ws) {
  int tid = threadIdx.x;
  if (tid < 4) {
    ws[576 + tid] = conv_b[tid];
    for (int n = 0; n < 16; ++n) {
      float acc = 0.f;
      for (int w = 0; w < 4; ++w)
        acc += conv_w[tid*4 + w] * (((n >> (3 - w)) & 1) ? -1.f : 1.f);
      ws[512 + n*4 + tid] = acc;               // transposed: Wt[n][o]
    }
  }
  if (tid < 16) {
    // column tid of U = circuit applied to basis vector e_tid
    float sr[16], si[16];
    for (int i = 0; i < 16; ++i) { sr[i] = (i == tid) ? 1.f : 0.f; si[i] = 0.f; }
    apply1q (sr, si, gRX(theta[0]), 0);
    apply1q (sr, si, gRY(theta[1]), 1);
    apply1q (sr, si, gRZ(theta[2]), 2);
    applyCtrl(sr, si, gRX(theta[3]), 0, 1);
    apply1q (sr, si, gRX(theta[4]), 3);
    applyCtrl(sr, si, gRY(theta[5]), 1, 2);
    apply1q (sr, si, gRZ(theta[6]), 0);
    applyCtrl(sr, si, gRZ(theta[7]), 2, 3);
    for (int n = 0; n < 16; ++n) { ws[n*16 + tid] = sr[n]; ws[256 + n*16 + tid] = si[n]; }
  }
}

// ---------------- main: one wave32 = 16 patches; D[n_basis, m_patch] = U * Psi^T -----------
__global__ __launch_bounds__(256) void quanv_main(const float* __restrict__ x,
                                                  const float* __restrict__ ws,
                                                  float* __restrict__ out,
                                                  int numTiles) {
  const float* Ur   = ws;
  const float* Ui   = ws + 256;
  const float* Wt   = ws + 512;    // Wt[n][o], 16x4
  const float* bias = ws + 576;

  const int lane = threadIdx.x & 31;
  const int half = lane >> 4;          // K sub-chunk / D row-half selector
  const int n    = lane & 15;          // A-row (basis) index == B/D column (patch) index
  const int wave   = (blockIdx.x * blockDim.x + threadIdx.x) >> 5;
  const int nWaves = (gridDim.x * blockDim.x) >> 5;

  // Loop-invariant A operand = U rows: lane supplies A[M=n, K=ch*4+2*half+{0,1}]
  v2f Are[4], Aim[4];
#pragma unroll
  for (int ch = 0; ch < 4; ++ch) {
    int k0 = ch*4 + half*2;
    Are[ch] = v2f{ Ur[n*16 + k0], Ur[n*16 + k0 + 1] };
    Aim[ch] = v2f{ Ui[n*16 + k0], Ui[n*16 + k0 + 1] };
  }
  // Per-lane reduction weights: this lane sees basis rows n = j + 8*half in D[j]
  float4 wv[8];
#pragma unroll
  for (int j = 0; j < 8; ++j)
    wv[j] = *(const float4*)(Wt + (j + 8*half)*4);
  const float bA = half ? bias[2] : bias[0];
  const float bB = half ? bias[3] : bias[1];

  for (int tile = wave; tile < numTiles; tile += nWaves) {
    // lane owns patch m = n of this tile (both halves load same patch, different K rows)
    int pt  = tile*16 + n;
    int img = pt / 196;
    int rem = pt - img*196;
    int pr  = rem / 14;
    int pc  = rem - pr*14;
    const float* px = x + img*784 + pr*56 + pc*2;
    float2 r0 = *(const float2*)(px);        // pixels (r,c),(r,c+1)
    float2 r1 = *(const float2*)(px + 28);   // pixels (r+1,c),(r+1,c+1)

    float c0 = __cosf(0.5f*r0.x), s0 = __sinf(0.5f*r0.x);
    float c1 = __cosf(0.5f*r0.y), s1 = __sinf(0.5f*r0.y);
    float c2 = __cosf(0.5f*r1.x), s2 = __sinf(0.5f*r1.x);
    float c3 = __cosf(0.5f*r1.y), s3 = __sinf(0.5f*r1.y);
    // psi[k] = P[k>>2] * Q[k&3]; B operand rows K = 4*ch + 2*half + {0,1}, column = patch
    float P[4] = { c0*c1, c0*s1, s0*c1, s0*s1 };
    float qa = half ? (s2*c3) : (c2*c3);     // Q[2*half + 0]
    float qb = half ? (s2*s3) : (c2*s3);     // Q[2*half + 1]

    v8f Dre = {0.f,0.f,0.f,0.f,0.f,0.f,0.f,0.f};
    v8f Dim = {0.f,0.f,0.f,0.f,0.f,0.f,0.f,0.f};
#pragma unroll
    for (int ch = 0; ch < 4; ++ch) {
      v2f bb = v2f{ P[ch]*qa, P[ch]*qb };
      // D[M=basis, N=patch] += U[basis, k-chunk] * Psi^T[k-chunk, patch]
      Dre = __builtin_amdgcn_wmma_f32_16x16x4_f32(false, Are[ch], false, bb, (short)0, Dre, false, false);
      Dim = __builtin_amdgcn_wmma_f32_16x16x4_f32(false, Aim[ch], false, bb, (short)0, Dim, false, false);
    }

    // In-lane reduction over basis rows held by this lane (n = j + 8*half)
    float a0 = 0.f, a1 = 0.f, a2 = 0.f, a3 = 0.f;
#pragma unroll
    for (int j = 0; j < 8; ++j) {
      float p = Dre[j]*Dre[j] + Dim[j]*Dim[j];
      a0 = fmaf(wv[j].x, p, a0);
      a1 = fmaf(wv[j].y, p, a1);
      a2 = fmaf(wv[j].z, p, a2);
      a3 = fmaf(wv[j].w, p, a3);
    }
    // combine the two 8-row halves: single cross-half butterfly per output
    a0 += __shfl_xor(a0, 16);
    a1 += __shfl_xor(a1, 16);
    a2 += __shfl_xor(a2, 16);
    a3 += __shfl_xor(a3, 16);

    // half-0 lanes write outputs o=0,1 ; half-1 lanes write o=2,3 (no EXEC divergence)
    float vA = (half ? a2 : a0) + bA;
    float vB = (half ? a3 : a1) + bB;
    int base = img*784 + half*392 + rem;     // out[b, 2*half, patch]
    out[base]       = vA;
    out[base + 196] = vB;
  }
}

extern "C" void kernel_launch(void* const* d_in, const int* in_sizes, int n_in,
                              void* d_out, int out_size, void* d_ws, size_t ws_size,
                              hipStream_t stream) {
  const float* x      = (const float*)d_in[0];   // [B,28,28] f32
  const float* theta  = (const float*)d_in[1];   // [8]
  const float* conv_w = (const float*)d_in[2];   // [4,4]
  const float* conv_b = (const float*)d_in[3];   // [4]
  float* out = (float*)d_out;
  float* ws  = (float*)d_ws;

  int B = in_sizes[0] / 784;
  int numPatches = B * 196;                      // 802816 for B=4096
  int numTiles   = numPatches / 16;              // exactly divisible for B=4096

  quanv_precompute<<<1, 32, 0, stream>>>(theta, conv_w, conv_b, ws);
  quanv_main<<<1024, 256, 0, stream>>>(x, ws, out, numTiles);
}